// DelayBuffer_85581518340253
// MI455X (gfx1250) — compile-verified
//
#include <hip/hip_runtime.h>

// DelayBuffer: out[b, t, di*1024 + c] = in[b, (t >= d) ? t-d : t, c],
// d = 1<<di for di in 0..5  (DELAYS = 1,2,4,8,16,32).
// Shapes fixed by the reference: B=4, S=4096, D=1024, f32 in/out.
//
// Pure streaming copy: HBM floor ~ (64 MiB read + 384 MiB write) / 23.3 TB/s
// ~= 20 us. Strategy: float4 (b128) memory ops, non-temporal stores so the
// 384 MiB output stream does not evict the 64 MiB input from the 192 MB L2
// (the input is logically read 6x; keeping it L2-resident makes those re-reads
// free of HBM cost). All row indexing is wave-uniform -> SALU.

typedef float v4f __attribute__((ext_vector_type(4)));

#define S_DIM 4096
#define D4    256   // 1024 floats = 256 float4
#define NDEL  6

__global__ __launch_bounds__(256) void delay_concat_kernel(
    const v4f* __restrict__ in, v4f* __restrict__ out)
{
    const int c = threadIdx.x;        // float4 column within D: 0..255
    const int t = blockIdx.x;         // time step: 0..4095
    const int b = blockIdx.y;         // batch: 0..3

    const size_t in_row_base = (size_t)b * S_DIM;                 // row units
    const size_t out_base =
        (((size_t)b * S_DIM + (size_t)t) * NDEL) * D4 + (size_t)c;

    // Gather the 6 source rows first (compiler can clause the 6 b128 loads),
    // then stream 6 NT stores.
    v4f vals[NDEL];
#pragma unroll
    for (int di = 0; di < NDEL; ++di) {
        const int d  = 1 << di;                     // 1,2,4,8,16,32
        const int st = (t >= d) ? (t - d) : t;      // wave-uniform select
        vals[di] = in[(in_row_base + (size_t)st) * D4 + (size_t)c];
    }
#pragma unroll
    for (int di = 0; di < NDEL; ++di) {
        __builtin_nontemporal_store(vals[di], &out[out_base + (size_t)di * D4]);
    }
}

extern "C" void kernel_launch(void* const* d_in, const int* in_sizes, int n_in,
                              void* d_out, int out_size, void* d_ws, size_t ws_size,
                              hipStream_t stream) {
    (void)in_sizes; (void)n_in; (void)out_size; (void)d_ws; (void)ws_size;

    const v4f* in  = (const v4f*)d_in[0];   // [4, 4096, 1024] f32
    v4f*       out = (v4f*)d_out;           // [4, 4096, 6144] f32

    // One block per (b, t) row; 256 threads = 256 float4 columns = 8 wave32s.
    dim3 grid(S_DIM, 4, 1);
    dim3 block(256, 1, 1);
    hipLaunchKernelGGL(delay_concat_kernel, grid, block, 0, stream, in, out);
}